// CRFLoss_15040975470979
// MI455X (gfx1250) — compile-verified
//
#include <hip/hip_runtime.h>

// CRF loss forward for MI455X (gfx1250, wave32).
// scores: (512,128,48,48) f32, target: (512,128) int, mask: (512,128) bool.
// One workgroup per batch element; 512-step sequential log-semiring scan with
// LDS double buffering. Column reductions of the exp() matrix use exact-f32
// V_WMMA_F32_16X16X4_F32 with an all-ones A matrix (D[m][j] = colsum_j + C).
// Next-step tile prefetch uses the Tensor Data Mover (tensor_load_to_lds,
// TENSORcnt) when available; falls back to per-lane async-to-LDS (ASYNCcnt),
// then to a register-staged pipeline.

typedef float v2f  __attribute__((ext_vector_type(2)));
typedef float v8f  __attribute__((ext_vector_type(8)));
typedef unsigned int u32x4 __attribute__((ext_vector_type(4)));
typedef int   i32x4 __attribute__((ext_vector_type(4)));
typedef int   i32x8 __attribute__((ext_vector_type(8)));

#if __has_builtin(__builtin_amdgcn_tensor_load_to_lds) && \
    __has_builtin(__builtin_amdgcn_s_wait_tensorcnt)
#define CRF_USE_TDM 1
#elif __has_builtin(__builtin_amdgcn_global_load_async_to_lds_b128) && \
      __has_builtin(__builtin_amdgcn_s_wait_asynccnt)
#define CRF_USE_ASYNC 1
#endif

namespace {
constexpr int S  = 512;
constexpr int B  = 128;
constexpr int T  = 48;
constexpr int TT = T * T;        // 2304 floats per (s,b) tile = 9216 B
constexpr int START_TAG = 46;
constexpr int END_TAG   = 47;
constexpr int NTH = 256;         // 8 wave32
constexpr int V4  = TT / 4;      // 576 float4 per tile
}

#ifdef CRF_USE_TDM
// Issue one TDM transfer of a full 9216-B step tile: global -> LDS.
// D# descriptor: 1-D tile, count=1, data_size=4B (code 2), no gather /
// iterate / pad / multicast; tensor_dim0 = tile_dim0 = stride0 = TT elements.
__device__ __forceinline__ void tdm_load_tile(const float* gsrc, float* ldst) {
    const unsigned long long ga = (unsigned long long)(size_t)gsrc;
    const unsigned lds_off = (unsigned)(size_t)ldst;   // low 32 bits = LDS byte offset
    u32x4 g0;
    g0[0] = 1u;                                        // count=1, user mode, no gather
    g0[1] = lds_off;                                   // lds_addr[31:0]
    g0[2] = (unsigned)ga;                              // global_addr[31:0]
    g0[3] = (unsigned)((ga >> 32) & 0x01FFFFFFu)       // global_addr[56:32]
            | 0x80000000u;                             // type = 2 (bits 127:126)
    i32x8 g1;
    g1[0] = 0x20000;                                   // data_size=2 (4B), wg_mask=0
    g1[1] = (int)((unsigned)(TT & 0xFFFF) << 16);      // tensor_dim0[15:0] @ bits 63:48
    g1[2] = (int)(1u << 16);                           // tensor_dim0[31:16]=0 | tensor_dim1=1
    g1[3] = (int)((unsigned)(TT & 0xFFFF) << 16);      // tile_dim0 = TT @ bits 127:112
    g1[4] = 0;                                         // tile_dim1/2 unused
    g1[5] = (int)TT;                                   // tensor_dim0_stride[31:0]
    g1[6] = 0;
    g1[7] = 0;
    const i32x4 z4 = {};
#if __clang_major__ >= 23
    const i32x8 z8 = {};
    __builtin_amdgcn_tensor_load_to_lds(g0, g1, z4, z4, z8, 0);
#else
    __builtin_amdgcn_tensor_load_to_lds(g0, g1, z4, z4, 0);
#endif
}
#endif

#ifdef CRF_USE_ASYNC
__device__ __forceinline__ void async_cp16(const float* g, float* l) {
    __builtin_amdgcn_global_load_async_to_lds_b128(
        (__attribute__((address_space(1))) void*)g,
        (__attribute__((address_space(3))) void*)l, 0, 0);
}
#endif

__global__ void crf_zero_out(float* out) { out[0] = 0.0f; }

__global__ __launch_bounds__(NTH, 2)
void crf_forward_kernel(const float* __restrict__ scores,
                        const int* __restrict__ target,
                        const unsigned char* __restrict__ mask,
                        float* __restrict__ out)
{
    __shared__ float tile[2][TT];   // double-buffered score tiles
    __shared__ float ebuf[TT];      // exp() matrix for current step
    __shared__ float part[T];       // partition vector (log domain)
    __shared__ float tg_sh;         // gathered target-energy for this block

    const int b    = blockIdx.x;
    const int tid  = threadIdx.x;
    const int lane = tid & 31;
    const int wave = tid >> 5;

    const float* gbase = scores + (size_t)b * TT;   // step stride = B*TT

    // ---- preload tile 0 ----
#ifdef CRF_USE_TDM
    if (wave == 0) {
        tdm_load_tile(gbase, tile[0]);
        __builtin_amdgcn_s_wait_tensorcnt(0);
    }
#elif defined(CRF_USE_ASYNC)
    async_cp16(gbase + 4 * tid,         tile[0] + 4 * tid);
    async_cp16(gbase + 4 * (tid + 256), tile[0] + 4 * (tid + 256));
    if (tid < V4 - 512)
        async_cp16(gbase + 4 * (tid + 512), tile[0] + 4 * (tid + 512));
    __builtin_amdgcn_s_wait_asynccnt(0);
#else
    {
        const float4* g = (const float4*)gbase;
        float4* l = (float4*)tile[0];
        l[tid]       = g[tid];
        l[tid + 256] = g[tid + 256];
        if (tid < V4 - 512) l[tid + 512] = g[tid + 512];
    }
#endif
    __syncthreads();

    float tg_acc = 0.0f;

    for (int s = 0; s < S; ++s) {
        float* cur = tile[s & 1];
        float* nxt = tile[(s + 1) & 1];
        const bool has_next = (s + 1 < S);

        // ---- kick off next-tile prefetch (overlaps this step's compute) ----
#ifdef CRF_USE_TDM
        if (wave == 0 && has_next)
            tdm_load_tile(gbase + (size_t)(s + 1) * B * TT, nxt);
#elif defined(CRF_USE_ASYNC)
        if (has_next) {
            const float* g = gbase + (size_t)(s + 1) * B * TT;
            async_cp16(g + 4 * tid,         nxt + 4 * tid);
            async_cp16(g + 4 * (tid + 256), nxt + 4 * (tid + 256));
            if (tid < V4 - 512)
                async_cp16(g + 4 * (tid + 512), nxt + 4 * (tid + 512));
        }
#else
        float4 pf0 = make_float4(0, 0, 0, 0);
        float4 pf1 = make_float4(0, 0, 0, 0);
        float4 pf2 = make_float4(0, 0, 0, 0);
        if (has_next) {
            const float4* g = (const float4*)(gbase + (size_t)(s + 1) * B * TT);
            pf0 = g[tid];
            pf1 = g[tid + 256];
            if (tid < V4 - 512) pf2 = g[tid + 512];
        }
#endif

        const unsigned char mk = mask[(size_t)s * B + b];
        float mx = 0.0f;

        if (s == 0) {
            // part0[j] = scores[0, b, START_TAG, j]
            if (tid < T) part[tid] = cur[START_TAG * T + tid];
        } else if (mk) {
            // mx = max_i part[i], computed redundantly per wave (no barrier).
            float v = part[lane];
            if (lane < T - 32) v = fmaxf(v, part[32 + lane]);
            #pragma unroll
            for (int d = 16; d; d >>= 1) v = fmaxf(v, __shfl_xor(v, d, 32));
            mx = v;

            // E[i][j] = exp(cur[i][j] + part[i] - mx); 9 elements per thread.
            for (int idx = tid; idx < TT; idx += NTH) {
                const int i = idx / T;
                ebuf[idx] = __expf(cur[idx] + part[i] - mx);
            }
        }
        __syncthreads();   // ebuf ready; all reads of `part` complete

        if (s > 0 && mk && wave < 3) {
            // colsum[j] = sum_i E[i][j] for this wave's 16 columns, via
            // D = ones(16x4) x E_chunk(4x16) + C  (exact f32 accumulate).
            // Assumed B layout (4x16 f32, 2 VGPRs): lane L, elem v holds
            // K = v + 2*(L>=16), N = L&15  (transpose of documented 16x4 A).
            const int colbase = wave * 16;
            const int col     = colbase + (lane & 15);
            const int khalf   = (lane >> 4) << 1;   // 0 or 2
            v2f a; a[0] = 1.0f; a[1] = 1.0f;        // all-ones A matrix

            // Stage all 24 LDS reads first (12 ds_load_2addr_b32) so their
            // latency overlaps, instead of a load->wait->wmma serial chain.
            v2f bv[12];
            #pragma unroll
            for (int c = 0; c < 12; ++c) {
                const int krow = c * 4 + khalf;
                bv[c][0] = ebuf[(krow + 0) * T + col];
                bv[c][1] = ebuf[(krow + 1) * T + col];
            }

            // 4 independent accumulator chains (dependency depth 3).
            v8f acc0 = {}, acc1 = {}, acc2 = {}, acc3 = {};
            #pragma unroll
            for (int c = 0; c < 12; c += 4) {
                acc0 = __builtin_amdgcn_wmma_f32_16x16x4_f32(
                           false, a, false, bv[c + 0], (short)0, acc0, false, false);
                acc1 = __builtin_amdgcn_wmma_f32_16x16x4_f32(
                           false, a, false, bv[c + 1], (short)0, acc1, false, false);
                acc2 = __builtin_amdgcn_wmma_f32_16x16x4_f32(
                           false, a, false, bv[c + 2], (short)0, acc2, false, false);
                acc3 = __builtin_amdgcn_wmma_f32_16x16x4_f32(
                           false, a, false, bv[c + 3], (short)0, acc3, false, false);
            }
            // D row M=0 lives in elem 0 of lanes 0-15 (N = lane).
            if (lane < 16) {
                const float cs = acc0[0] + acc1[0] + acc2[0] + acc3[0];
                part[col] = mx + __logf(cs);
            }
        }

        // Fused target-energy gather from the resident tile (wave 7 is idle
        // during the WMMA phase).
        if (wave == 7 && lane == 0 && mk) {
            const int t = target[(size_t)s * B + b];
            tg_acc += cur[t];
        }

        // ---- ensure next tile is resident before the step boundary ----
#ifdef CRF_USE_TDM
        if (wave == 0 && has_next)
            __builtin_amdgcn_s_wait_tensorcnt(0);
#elif defined(CRF_USE_ASYNC)
        if (has_next)
            __builtin_amdgcn_s_wait_asynccnt(0);
#else
        if (has_next) {
            float4* l = (float4*)nxt;
            l[tid]       = pf0;
            l[tid + 256] = pf1;
            if (tid < V4 - 512) l[tid + 512] = pf2;
        }
#endif
        __syncthreads();   // part updated + next tile resident
    }

    if (wave == 7 && lane == 0) tg_sh = tg_acc;
    __syncthreads();
    if (tid == 0) {
        // loss contribution of this batch element: (part[END] - tg) / B
        const float v = (part[END_TAG] - tg_sh) * (1.0f / (float)B);
        atomicAdd(out, v);
    }
}

extern "C" void kernel_launch(void* const* d_in, const int* in_sizes, int n_in,
                              void* d_out, int out_size, void* d_ws, size_t ws_size,
                              hipStream_t stream) {
    const float*         scores = (const float*)d_in[0];
    const int*           target = (const int*)d_in[1];
    const unsigned char* mask   = (const unsigned char*)d_in[2];
    float*               out    = (float*)d_out;

    crf_zero_out<<<1, 1, 0, stream>>>(out);
    crf_forward_kernel<<<B, NTH, 0, stream>>>(scores, target, mask, out);
}